// InterpretableMultiHeadAttention_19481971654730
// MI455X (gfx1250) — compile-verified
//
#include <hip/hip_runtime.h>

typedef __attribute__((ext_vector_type(2))) float v2f;
typedef __attribute__((ext_vector_type(8))) float v8f;

#define BATCH 32
#define SEQ 512
#define DMODEL 256
#define NH 4
#define DKD 64
#define NEG_INF (-__builtin_inff())

__device__ __forceinline__ v8f wmma4(v2f a, v2f b, v8f c) {
  // D = A(16x4) * B(4x16) + C(16x16), all f32
  return __builtin_amdgcn_wmma_f32_16x16x4_f32(false, a, false, b, (short)0, c,
                                               false, false);
}

__global__ void zero_kernel(float* __restrict__ p, int n) {
  int i = blockIdx.x * blockDim.x + threadIdx.x;
  if (i < n) p[i] = 0.0f;
}

// Generic projection GEMM: Y[m,n] = sum_k X[m,k] * W[n,k] + bias[n]
// Block = 128 threads (4 waves); each wave owns one 16-wide N tile.
template <int K>
__global__ __launch_bounds__(128) void proj_kernel(const float* __restrict__ X,
                                                   const float* __restrict__ W,
                                                   const float* __restrict__ bias,
                                                   float* __restrict__ Y, int N) {
  constexpr int LDA = K + 1;  // pad to avoid LDS bank conflicts
  __shared__ float ldsA[16 * LDA];
  const int tid = threadIdx.x;
  const int lane = tid & 31;
  const int wave = tid >> 5;
  const int m0 = blockIdx.x * 16;
  const int n0 = blockIdx.y * 64 + wave * 16;

  for (int i = tid; i < 16 * K; i += 128) {
    int r = i / K, c = i % K;
    ldsA[r * LDA + c] = X[(size_t)(m0 + r) * K + c];
  }
  __syncthreads();

  const int l15 = lane & 15;
  const int koff = (lane >> 4) * 2;  // 0 or 2
  v8f acc;
  float bval = bias ? bias[n0 + l15] : 0.0f;
#pragma unroll
  for (int i = 0; i < 8; ++i) acc[i] = bval;

  const float* aRow = ldsA + l15 * LDA;
  const float* wRow = W + (size_t)(n0 + l15) * K;
  for (int k0 = 0; k0 < K; k0 += 4) {
    v2f a, b;
    a.x = aRow[k0 + koff];
    a.y = aRow[k0 + koff + 1];
    b.x = wRow[k0 + koff];
    b.y = wRow[k0 + koff + 1];
    acc = wmma4(a, b, acc);
  }

  const int rb = (lane >> 4) * 8;
#pragma unroll
  for (int i = 0; i < 8; ++i)
    Y[(size_t)(m0 + rb + i) * N + n0 + l15] = acc[i];
}

// One wave per (b, h, 16-query-row tile): scores -> causal softmax -> attn out
// (already transposed to [b,s,h,t]) -> A*V, mean-accumulated into heads ws.
__global__ __launch_bounds__(32) void attn_kernel(const float* __restrict__ qs,
                                                  const float* __restrict__ ks,
                                                  const float* __restrict__ vs,
                                                  float* __restrict__ heads,
                                                  float* __restrict__ attn) {
  constexpr int LS = SEQ + 4;   // score-row stride (bank-conflict pad)
  constexpr int LQ = DKD + 1;
  __shared__ float sQ[16 * LQ];
  __shared__ float sS[16 * LS];

  const int lane = threadIdx.x;
  const int qt = blockIdx.x, h = blockIdx.y, b = blockIdx.z;
  const int s0 = qt * 16;
  const int l15 = lane & 15;
  const int koff = (lane >> 4) * 2;
  const int rb = (lane >> 4) * 8;

  // Stage this head's Q tile: 16 x 64
  for (int i = lane; i < 16 * DKD; i += 32) {
    int r = i >> 6, c = i & 63;
    sQ[r * LQ + c] = qs[((size_t)b * SEQ + s0 + r) * DMODEL + h * DKD + c];
  }
  __syncthreads();

  // Scores: Q (16x64) . K^T tiles, scale 1/sqrt(64), causal mask to -inf
  const float* ksb = ks + (size_t)b * SEQ * DMODEL + h * DKD;
  const int ktmax = qt + 1;  // tiles with t0 <= s0+15
  for (int kt = 0; kt < SEQ / 16; ++kt) {
    const int t0 = kt * 16;
    if (kt < ktmax) {
      v8f acc;
#pragma unroll
      for (int i = 0; i < 8; ++i) acc[i] = 0.0f;
      const float* kRow = ksb + (size_t)(t0 + l15) * DMODEL;
#pragma unroll
      for (int k0 = 0; k0 < DKD; k0 += 4) {
        v2f a, bb;
        a.x = sQ[l15 * LQ + k0 + koff];
        a.y = sQ[l15 * LQ + k0 + koff + 1];
        bb.x = kRow[k0 + koff];
        bb.y = kRow[k0 + koff + 1];
        acc = wmma4(a, bb, acc);
      }
#pragma unroll
      for (int i = 0; i < 8; ++i) {
        int r = rb + i;
        sS[r * LS + t0 + l15] =
            (t0 + l15 <= s0 + r) ? acc[i] * 0.125f : NEG_INF;
      }
    } else {
#pragma unroll
      for (int i = 0; i < 8; ++i) sS[(rb + i) * LS + t0 + l15] = NEG_INF;
    }
  }
  __syncthreads();

  // Softmax: two lanes per row (halves of 256 cols), combined via shfl_xor 16
  {
    float* rowp = sS + l15 * LS + (lane >> 4) * 256;
    float m = NEG_INF;
    for (int c = 0; c < 256; ++c) m = fmaxf(m, rowp[c]);
    m = fmaxf(m, __shfl_xor(m, 16, 32));
    float sum = 0.0f;
    for (int c = 0; c < 256; ++c) {
      float p = __expf(rowp[c] - m);
      rowp[c] = p;
      sum += p;
    }
    sum += __shfl_xor(sum, 16, 32);
    float inv = 1.0f / sum;
    for (int c = 0; c < 256; ++c) rowp[c] *= inv;
  }
  __syncthreads();

  // Write attn in transposed layout [b, s, h, t] (coalesced 128B stores)
  float* abase = attn + (((size_t)b * SEQ + s0) * NH + h) * SEQ;
  for (int it = 0; it < 16 * SEQ / 32; ++it) {
    int idx = it * 32 + lane;
    int r = idx >> 9, c = idx & (SEQ - 1);
    abase[(size_t)r * NH * SEQ + c] = sS[r * LS + c];
  }

  // A*V: (16 x kmax) . (kmax x 64), causal truncation; mean-accumulate 0.25x
  const float* vb = vs + (size_t)b * SEQ * DKD;
  const int kmax = s0 + 16;
  for (int nt = 0; nt < 4; ++nt) {
    const int n0 = nt * 16;
    v8f acc;
#pragma unroll
    for (int i = 0; i < 8; ++i) acc[i] = 0.0f;
    for (int k0 = 0; k0 < kmax; k0 += 4) {
      v2f a, bb;
      a.x = sS[l15 * LS + k0 + koff];
      a.y = sS[l15 * LS + k0 + koff + 1];
      bb.x = vb[(size_t)(k0 + koff) * DKD + n0 + l15];
      bb.y = vb[(size_t)(k0 + koff + 1) * DKD + n0 + l15];
      acc = wmma4(a, bb, acc);
    }
#pragma unroll
    for (int i = 0; i < 8; ++i) {
      atomicAdd(&heads[((size_t)b * SEQ + s0 + rb + i) * DKD + n0 + l15],
                0.25f * acc[i]);
    }
  }
}

extern "C" void kernel_launch(void* const* d_in, const int* in_sizes, int n_in,
                              void* d_out, int out_size, void* d_ws,
                              size_t ws_size, hipStream_t stream) {
  (void)in_sizes; (void)n_in; (void)out_size; (void)ws_size;
  const float* q  = (const float*)d_in[0];
  const float* k  = (const float*)d_in[1];
  const float* v  = (const float*)d_in[2];
  // d_in[3] = causal mask (bool) -> implemented analytically, unused
  const float* Wq = (const float*)d_in[4];   // [H,DK,D] -> rows [256,256]
  const float* bq = (const float*)d_in[5];   // [H,DK]   -> [256]
  const float* Wk = (const float*)d_in[6];
  const float* bk = (const float*)d_in[7];
  const float* Wv = (const float*)d_in[8];   // [64,256]
  const float* bv = (const float*)d_in[9];   // [64]
  const float* Wh = (const float*)d_in[10];  // [256,64]

  float* out  = (float*)d_out;                          // [B,S,D]
  float* attn = out + (size_t)BATCH * SEQ * DMODEL;     // [B,S,H,S]

  float* ws    = (float*)d_ws;
  float* qsb   = ws;                                    // [B,S,256]
  float* ksb   = qsb + (size_t)BATCH * SEQ * DMODEL;    // [B,S,256]
  float* vsb   = ksb + (size_t)BATCH * SEQ * DMODEL;    // [B,S,64]
  float* heads = vsb + (size_t)BATCH * SEQ * DKD;       // [B,S,64]

  const int M = BATCH * SEQ;  // 16384

  {
    int n = M * DKD;
    zero_kernel<<<(n + 255) / 256, 256, 0, stream>>>(heads, n);
  }
  proj_kernel<DMODEL><<<dim3(M / 16, (NH * DKD) / 64), 128, 0, stream>>>(
      q, Wq, bq, qsb, NH * DKD);
  proj_kernel<DMODEL><<<dim3(M / 16, (NH * DKD) / 64), 128, 0, stream>>>(
      k, Wk, bk, ksb, NH * DKD);
  proj_kernel<DMODEL><<<dim3(M / 16, 1), 128, 0, stream>>>(v, Wv, bv, vsb, DKD);
  attn_kernel<<<dim3(SEQ / 16, NH, BATCH), 32, 0, stream>>>(qsb, ksb, vsb,
                                                            heads, attn);
  proj_kernel<DKD><<<dim3(M / 16, DMODEL / 64), 128, 0, stream>>>(
      heads, Wh, nullptr, out, DMODEL);
}